// WPHOp_11776800326118
// MI455X (gfx1250) — compile-verified
//
#include <hip/hip_runtime.h>
#include <math.h>

// ---------------------------------------------------------------------------
// WPH statistics on gfx1250 via tensor-core (WMMA) FFTs.
//   fft2(x)  = F x F      (F[k][n] = exp(-2i*pi*k*n/128), symmetric)
//   ifft2(Y) = G Y G      (G[k][n] = exp(+2i*pi*k*n/128)/128)
// Complex GEMM = 4 real 128x128x128 GEMMs on V_WMMA_F32_16X16X4_F32.
// ---------------------------------------------------------------------------

#define MDIM 128
#define NDIM 128
#define MN   16384
#define BATCH 4
#define JSC  5
#define LANG 8
#define NPSI 40            // J*L
#define NFLD 45            // NPSI + J low-pass fields
#define KMOM 280
#define SMOM 20
#define OUTSTRIDE 580      // 2*KMOM + SMOM

typedef __attribute__((ext_vector_type(2))) float v2f;
typedef __attribute__((ext_vector_type(8))) float v8f;

// ------------------------- twiddle generation ------------------------------
__global__ __launch_bounds__(256)
void gen_dft(float* __restrict__ Fr, float* __restrict__ Fi,
             float* __restrict__ Gr, float* __restrict__ Gi,
             float* __restrict__ zbuf) {
  int i = blockIdx.x * 256 + threadIdx.x;          // 0..16383
  int k = i >> 7, n = i & 127;
  int m = (k * n) & 127;
  float th = (float)m * (6.283185307179586f / 128.0f);
  float c = __cosf(th), s = __sinf(th);
  Fr[i] = c;            Fi[i] = -s;                // exp(-i th)
  Gr[i] = c * (1.0f/128.0f); Gi[i] = s * (1.0f/128.0f); // exp(+i th)/128
  zbuf[i] = 0.0f;
}

// ------------------- batched complex GEMM (128x128x128) --------------------
// C = A * B, all 128x128 row-major, complex split into (r,i) planes.
// One wave32 computes one 16x16 tile of C using V_WMMA_F32_16X16X4_F32.
// Per-operand batch strides allow broadcasting twiddle matrices / zero imag.
__global__ __launch_bounds__(256)
void cgemm128(const float* __restrict__ Ar, const float* __restrict__ Ai,
              const float* __restrict__ Br, const float* __restrict__ Bi,
              float* __restrict__ Cr, float* __restrict__ Ci,
              int asr, int asi, int bsr, int bsi, int cs) {
  const long batch = blockIdx.y;
  const int wave = threadIdx.x >> 5;
  const int lane = threadIdx.x & 31;
  const int tile = blockIdx.x * 8 + wave;          // 0..63
  const int tm = (tile >> 3) << 4;                 // row base of C tile
  const int tn = (tile & 7) << 4;                  // col base of C tile

  Ar += batch * asr;  Ai += batch * asi;
  Br += batch * bsr;  Bi += batch * bsi;
  Cr += batch * cs;   Ci += batch * cs;

  // A fragment (16x4): lane&15 = row M; half-wave selects K pair {0,1}/{2,3}
  const int am = tm + (lane & 15);
  const int ak = (lane >> 4) << 1;
  // B fragment (4x16): lane&15 = col N; half-wave selects K pair {0,1}/{2,3}
  const int bn = tn + (lane & 15);
  const int bk = (lane >> 4) << 1;

  v8f accRR = {}, accII = {}, accRI = {}, accIR = {};

  for (int k = 0; k < 128; k += 4) {
    const float* pAr = Ar + am * 128 + (k + ak);
    const float* pAi = Ai + am * 128 + (k + ak);
    const float* pBr = Br + (k + bk) * 128 + bn;
    const float* pBi = Bi + (k + bk) * 128 + bn;
    v2f ar, ai, br, bi;
    ar.x = pAr[0];   ar.y = pAr[1];
    ai.x = pAi[0];   ai.y = pAi[1];
    br.x = pBr[0];   br.y = pBr[128];
    bi.x = pBi[0];   bi.y = pBi[128];
    accRR = __builtin_amdgcn_wmma_f32_16x16x4_f32(false, ar, false, br, (short)0, accRR, false, false);
    accII = __builtin_amdgcn_wmma_f32_16x16x4_f32(false, ai, false, bi, (short)0, accII, false, false);
    accRI = __builtin_amdgcn_wmma_f32_16x16x4_f32(false, ar, false, bi, (short)0, accRI, false, false);
    accIR = __builtin_amdgcn_wmma_f32_16x16x4_f32(false, ai, false, br, (short)0, accIR, false, false);
  }

  // C/D layout: VGPR r -> rows tm+r (lanes 0-15) / tm+8+r (lanes 16-31)
  const int cm = tm + ((lane >> 4) << 3);
  const int cn = tn + (lane & 15);
  #pragma unroll
  for (int r = 0; r < 8; ++r) {
    Cr[(cm + r) * 128 + cn] = accRR[r] - accII[r];
    Ci[(cm + r) * 128 + cn] = accRI[r] + accIR[r];
  }
}

// ------------------- filter multiply in Fourier space ----------------------
// Y[b,f] = xf[b] * filt[f]   (filt real; f<40 -> psi_f[f], else phi_f[f-40])
__global__ __launch_bounds__(256)
void apply_filters(const float* __restrict__ xfr, const float* __restrict__ xfi,
                   const float* __restrict__ psi, const float* __restrict__ phi,
                   float* __restrict__ Yr, float* __restrict__ Yi) {
  int bf = blockIdx.y;                  // 0..179
  int b = bf / NFLD, f = bf % NFLD;
  const float* filt = (f < NPSI) ? (psi + (long)f * MN) : (phi + (long)(f - NPSI) * MN);
  int i = blockIdx.x * 256 + threadIdx.x;       // grid.x = 64 -> covers 16384
  float s = filt[i];
  long o = (long)bf * MN + i;
  long xo = (long)b * MN + i;
  Yr[o] = xfr[xo] * s;
  Yi[o] = xfi[xo] * s;
}

// ------------------------- phase harmonic helper ---------------------------
// [z]^p = |z| * exp(i p angle(z)); p is a small non-negative integer here,
// so use exact binary powering of the unit phasor u = z/|z|.
__device__ __forceinline__ void phase_harm(float re, float im, float pf,
                                           float& hr, float& hi) {
  float mag = sqrtf(re * re + im * im);
  int p = (int)pf;
  if (p == 0) { hr = mag; hi = 0.0f; return; }
  if (p == 1) { hr = re;  hi = im;   return; }
  float inv = (mag > 0.0f) ? (1.0f / mag) : 0.0f;
  float br = re * inv, bi = im * inv;   // unit phasor
  float rr = 1.0f, ri = 0.0f;
  int e = p;
  while (e) {
    if (e & 1) { float t = rr * br - ri * bi; ri = rr * bi + ri * br; rr = t; }
    float t2 = br * br - bi * bi; bi = 2.0f * br * bi; br = t2;
    e >>= 1;
  }
  hr = mag * rr; hi = mag * ri;
}

// ------------------------- covariance reduction ----------------------------
// One block per (k, b): cov = mean(h1*h2t) - mean(h1)*mean(h2t), * weight.
__global__ __launch_bounds__(256)
void cov_kernel(const float* __restrict__ Wr, const float* __restrict__ Wi,
                const int* __restrict__ psi1, const int* __restrict__ psi2,
                const float* __restrict__ p1, const float* __restrict__ p2,
                const int* __restrict__ pseudo,
                const int* __restrict__ tx, const int* __restrict__ ty,
                const float* __restrict__ weights,
                float* __restrict__ out) {
  int k = blockIdx.x, b = blockIdx.y;
  int f1 = psi1[k], f2 = psi2[k];
  float P1 = p1[k], P2 = p2[k];
  int ps = pseudo[k];
  int sx = tx[k], sy = ty[k];
  const float* w1r = Wr + ((long)b * NFLD + f1) * MN;
  const float* w1i = Wi + ((long)b * NFLD + f1) * MN;
  const float* w2r = Wr + ((long)b * NFLD + f2) * MN;
  const float* w2i = Wi + ((long)b * NFLD + f2) * MN;

  float s1r = 0, s1i = 0, s2r = 0, s2i = 0, spr = 0, spi = 0;
  for (int i = threadIdx.x; i < MN; i += 256) {
    int y = i >> 7, x = i & 127;
    int i2 = (((y - sy) & 127) << 7) | ((x - sx) & 127);   // roll by (ty,tx)
    float h1r, h1i, h2r, h2i;
    phase_harm(w1r[i], w1i[i], P1, h1r, h1i);
    phase_harm(w2r[i2], w2i[i2], P2, h2r, h2i);
    if (ps <= 0) h2i = -h2i;                               // conj when !pseudo
    s1r += h1r; s1i += h1i;
    s2r += h2r; s2i += h2i;
    spr += h1r * h2r - h1i * h2i;
    spi += h1r * h2i + h1i * h2r;
  }

  __shared__ float red[6 * 256];
  int t = threadIdx.x;
  red[0 * 256 + t] = s1r; red[1 * 256 + t] = s1i;
  red[2 * 256 + t] = s2r; red[3 * 256 + t] = s2i;
  red[4 * 256 + t] = spr; red[5 * 256 + t] = spi;
  __syncthreads();
  for (int st = 128; st > 0; st >>= 1) {
    if (t < st)
      for (int c = 0; c < 6; ++c) red[c * 256 + t] += red[c * 256 + t + st];
    __syncthreads();
  }
  if (t == 0) {
    const float inv = 1.0f / (float)MN;
    float m1r = red[0] * inv, m1i = red[256] * inv;
    float m2r = red[512] * inv, m2i = red[768] * inv;
    float cr = red[1024] * inv - (m1r * m2r - m1i * m2i);
    float ci = red[1280] * inv - (m1r * m2i + m1i * m2r);
    float w = weights[k];
    out[(long)b * OUTSTRIDE + k]        = cr * w;
    out[(long)b * OUTSTRIDE + KMOM + k] = ci * w;
  }
}

// ------------------------- scaling moments ---------------------------------
// sm = mean(|hl|^2) - |mean(hl)|^2, hl = [x*phi_j]^p
__global__ __launch_bounds__(256)
void sm_kernel(const float* __restrict__ Wr, const float* __restrict__ Wi,
               const int* __restrict__ phi_j, const float* __restrict__ sm_p,
               float* __restrict__ out) {
  int s = blockIdx.x, b = blockIdx.y;
  int f = NPSI + phi_j[s];
  float P = sm_p[s];
  const float* lr = Wr + ((long)b * NFLD + f) * MN;
  const float* li = Wi + ((long)b * NFLD + f) * MN;

  float sr = 0, si = 0, s2 = 0;
  for (int i = threadIdx.x; i < MN; i += 256) {
    float hr, hi;
    phase_harm(lr[i], li[i], P, hr, hi);
    sr += hr; si += hi; s2 += hr * hr + hi * hi;
  }
  __shared__ float red[3 * 256];
  int t = threadIdx.x;
  red[t] = sr; red[256 + t] = si; red[512 + t] = s2;
  __syncthreads();
  for (int st = 128; st > 0; st >>= 1) {
    if (t < st)
      for (int c = 0; c < 3; ++c) red[c * 256 + t] += red[c * 256 + t + st];
    __syncthreads();
  }
  if (t == 0) {
    const float inv = 1.0f / (float)MN;
    float mr = red[0] * inv, mi = red[256] * inv;
    out[(long)b * OUTSTRIDE + 2 * KMOM + s] = red[512] * inv - (mr * mr + mi * mi);
  }
}

// ---------------------------------------------------------------------------
extern "C" void kernel_launch(void* const* d_in, const int* in_sizes, int n_in,
                              void* d_out, int out_size, void* d_ws, size_t ws_size,
                              hipStream_t stream) {
  const float* x     = (const float*)d_in[0];
  const float* psi_f = (const float*)d_in[1];
  const float* phi_f = (const float*)d_in[2];
  const int*   psi1  = (const int*)d_in[3];
  const int*   psi2  = (const int*)d_in[4];
  const float* p1    = (const float*)d_in[5];
  const float* p2    = (const float*)d_in[6];
  const int*   pseudo= (const int*)d_in[7];
  const int*   tx    = (const int*)d_in[8];
  const int*   ty    = (const int*)d_in[9];
  const float* wgt   = (const float*)d_in[10];
  const int*   phi_j = (const int*)d_in[11];
  const float* sm_p  = (const float*)d_in[12];
  float* out = (float*)d_out;

  // workspace layout (floats)
  float* ws = (float*)d_ws;
  const long FIELD = (long)BATCH * NFLD * MN;   // 2,949,120
  float* Fr  = ws;                 // 16384
  float* Fi  = Fr  + MN;
  float* Gr  = Fi  + MN;
  float* Gi  = Gr  + MN;
  float* zb  = Gi  + MN;           // zero imag plane
  float* Ur  = zb  + MN;           // BATCH*MN
  float* Ui  = Ur  + (long)BATCH * MN;
  float* xfr = Ui  + (long)BATCH * MN;
  float* xfi = xfr + (long)BATCH * MN;
  float* Yr  = xfi + (long)BATCH * MN;  // BATCH*NFLD*MN (also holds W at the end)
  float* Yi  = Yr  + FIELD;
  float* Vr  = Yi  + FIELD;
  float* Vi  = Vr  + FIELD;
  size_t need = (size_t)((Vi + FIELD) - ws) * sizeof(float);
  if (ws_size < need) return;      // not enough scratch; bail deterministically

  // 1. twiddle matrices + zero plane
  gen_dft<<<dim3(MN / 256), 256, 0, stream>>>(Fr, Fi, Gr, Gi, zb);

  // 2. forward FFT2: U = x*F ; xf = F*U     (x real -> imag plane = zeros)
  cgemm128<<<dim3(8, BATCH), 256, 0, stream>>>(x, zb, Fr, Fi, Ur, Ui,
                                               MN, 0, 0, 0, MN);
  cgemm128<<<dim3(8, BATCH), 256, 0, stream>>>(Fr, Fi, Ur, Ui, xfr, xfi,
                                               0, 0, MN, MN, MN);

  // 3. Y[b,f] = xf[b] * filter[f]
  apply_filters<<<dim3(64, BATCH * NFLD), 256, 0, stream>>>(xfr, xfi, psi_f, phi_f, Yr, Yi);

  // 4. inverse FFT2 of all 180 fields: V = Y*G ; W = G*V  (W reuses Y buffers)
  cgemm128<<<dim3(8, BATCH * NFLD), 256, 0, stream>>>(Yr, Yi, Gr, Gi, Vr, Vi,
                                                      MN, MN, 0, 0, MN);
  cgemm128<<<dim3(8, BATCH * NFLD), 256, 0, stream>>>(Gr, Gi, Vr, Vi, Yr, Yi,
                                                      0, 0, MN, MN, MN);

  // 5. moments
  cov_kernel<<<dim3(KMOM, BATCH), 256, 0, stream>>>(Yr, Yi, psi1, psi2, p1, p2,
                                                    pseudo, tx, ty, wgt, out);
  sm_kernel<<<dim3(SMOM, BATCH), 256, 0, stream>>>(Yr, Yi, phi_j, sm_p, out);
}